// GeoDynamicLayer_16106127360645
// MI455X (gfx1250) — compile-verified
//
#include <hip/hip_runtime.h>
#include <hip/hip_bf16.h>

// ---------------- problem constants (from reference) ----------------
#define IN_DIM   1024
#define OUT_DIM  2048
#define RANK     8
#define STEPS    4
#define BATCH    8
#define NSEQ     1024
#define MAXV     1.5f
#define CLAMPV   1e3f

typedef __attribute__((ext_vector_type(2))) float v2f;
typedef __attribute__((ext_vector_type(8))) float v8f;

// ---- CDNA5 async global->LDS copy (ASYNCcnt-tracked, bypasses VGPRs) ----
__device__ __forceinline__ void async_load_b128(unsigned lds_addr, const float* gaddr) {
    asm volatile("global_load_async_to_lds_b128 %0, %1, off"
                 :: "v"(lds_addr), "v"(gaddr) : "memory");
}
__device__ __forceinline__ void wait_async0() {
    asm volatile("s_wait_asynccnt 0x0" ::: "memory");
}
// low 32 bits of a generic pointer into LDS == LDS byte offset (ISA 10.2)
__device__ __forceinline__ unsigned lds_addr_of(const void* p) {
    return (unsigned)(uintptr_t)p;
}

// =====================================================================
// 1) z = mean(x, axis=1)   x:[B,N,IN] -> z:[B,IN]
// =====================================================================
__global__ void k_mean(const float* __restrict__ x, float* __restrict__ z) {
    int b = blockIdx.y;
    int i = blockIdx.x * blockDim.x + threadIdx.x;      // 0..IN-1
    const float* xp = x + (size_t)b * NSEQ * IN_DIM + i;
    float s = 0.f;
    #pragma unroll 4
    for (int n = 0; n < NSEQ; ++n) s += xp[(size_t)n * IN_DIM];
    z[b * IN_DIM + i] = s * (1.0f / NSEQ);
}

// =====================================================================
// 2) out[b,col] = z[b,:] @ W[:,col] + bias[col]    (M=8 GEMV, mem-bound)
//    W is ~200MB single-use -> non-temporal loads keep U/x hot in L2.
// =====================================================================
__global__ void k_zmatmul(const float* __restrict__ z, const float* __restrict__ W,
                          const float* __restrict__ bv, float* __restrict__ out,
                          int ncols) {
    int col = blockIdx.x * blockDim.x + threadIdx.x;
    int b   = blockIdx.y;
    if (col >= ncols) return;
    const float* zp = z + b * IN_DIM;
    float acc = bv[col];
    #pragma unroll 4
    for (int k = 0; k < IN_DIM; ++k) {
        float w = __builtin_nontemporal_load(&W[(size_t)k * ncols + col]);
        acc = fmaf(zp[k], w, acc);
    }
    out[(size_t)b * ncols + col] = acc;
}

// =====================================================================
// 3) velocity-clamp scale per batch, via 8x8 Gram matrices:
//    ||PQ^T - QP^T||^2 = 2*sum(PtP .* QtQ) - 2*tr((QtP)(QtP))
//    ||Bp Bq^T||^2     = sum(BptBp .* BqtBq)
// =====================================================================
__global__ void k_scale(const float* __restrict__ Pf, const float* __restrict__ Qf,
                        const float* __restrict__ Bpf, const float* __restrict__ Bqf,
                        float* __restrict__ sarr) {
    int b = blockIdx.x;
    int t = threadIdx.x;                 // 320 threads: 5 grams x 64 pairs
    __shared__ float gram[5][64];
    int g = t >> 6, p = t & 63, i = p >> 3, j = p & 7;
    const float* Pb  = Pf  + (size_t)b * OUT_DIM * RANK;
    const float* Qb  = Qf  + (size_t)b * OUT_DIM * RANK;
    const float* Bpb = Bpf + (size_t)b * (OUT_DIM - IN_DIM) * RANK;
    const float* Bqb = Bqf + (size_t)b * IN_DIM * RANK;
    float acc = 0.f;
    if (g == 0) {        // PtP
        for (int m = 0; m < OUT_DIM; ++m) acc = fmaf(Pb[m*RANK+i], Pb[m*RANK+j], acc);
    } else if (g == 1) { // QtQ
        for (int m = 0; m < OUT_DIM; ++m) acc = fmaf(Qb[m*RANK+i], Qb[m*RANK+j], acc);
    } else if (g == 2) { // QtP
        for (int m = 0; m < OUT_DIM; ++m) acc = fmaf(Qb[m*RANK+i], Pb[m*RANK+j], acc);
    } else if (g == 3) { // BptBp
        for (int m = 0; m < OUT_DIM - IN_DIM; ++m) acc = fmaf(Bpb[m*RANK+i], Bpb[m*RANK+j], acc);
    } else {             // BqtBq
        for (int m = 0; m < IN_DIM; ++m) acc = fmaf(Bqb[m*RANK+i], Bqb[m*RANK+j], acc);
    }
    gram[g][p] = acc;
    __syncthreads();
    if (t == 0) {
        float nM = 0.f, trMM = 0.f, nG = 0.f;
        for (int a = 0; a < 64; ++a) {
            nM = fmaf(gram[0][a], gram[1][a], nM);
            nG = fmaf(gram[3][a], gram[4][a], nG);
        }
        for (int ii = 0; ii < 8; ++ii)
            for (int jj = 0; jj < 8; ++jj)
                trMM = fmaf(gram[2][ii*8+jj], gram[2][jj*8+ii], trMM);
        float nA = 2.f * (nM - trMM);
        float cn = sqrtf(fmaxf(nA + nG, 0.f));
        sarr[b] = fminf(MAXV / (cn + 1e-6f), 1.0f);
    }
}

// =====================================================================
// 4) U[b] = U0  (broadcast copy, float4)
// =====================================================================
__global__ void k_initU(const float* __restrict__ U0, float* __restrict__ U) {
    int b = blockIdx.y;
    size_t i4 = (size_t)blockIdx.x * blockDim.x + threadIdx.x;   // float4 index
    const float4* src = (const float4*)U0;
    float4* dst = (float4*)(U + (size_t)b * OUT_DIM * IN_DIM);
    dst[i4] = src[i4];
}

// =====================================================================
// 5a) thin GEMM:  QtU[b,r,n] = sum_m Q[b,m,r] * U[b,m,n]   (and PtU)
// =====================================================================
__global__ void k_step_thin(const float* __restrict__ U,
                            const float* __restrict__ Pf, const float* __restrict__ Qf,
                            float* __restrict__ QtU, float* __restrict__ PtU) {
    int b = blockIdx.y;
    int n = blockIdx.x * 128 + threadIdx.x;
    __shared__ float qs[128][RANK];
    __shared__ float ps[128][RANK];
    const float* Ub = U  + (size_t)b * OUT_DIM * IN_DIM;
    const float* Pb = Pf + (size_t)b * OUT_DIM * RANK;
    const float* Qb = Qf + (size_t)b * OUT_DIM * RANK;
    float aq[RANK], ap[RANK];
    #pragma unroll
    for (int r = 0; r < RANK; ++r) { aq[r] = 0.f; ap[r] = 0.f; }
    for (int m0 = 0; m0 < OUT_DIM; m0 += 128) {
        for (int i = threadIdx.x; i < 128 * RANK; i += 128) {
            qs[i >> 3][i & 7] = Qb[m0 * RANK + i];
            ps[i >> 3][i & 7] = Pb[m0 * RANK + i];
        }
        __syncthreads();
        for (int mm = 0; mm < 128; ++mm) {
            float u = Ub[(size_t)(m0 + mm) * IN_DIM + n];
            #pragma unroll
            for (int r = 0; r < RANK; ++r) {
                aq[r] = fmaf(qs[mm][r], u, aq[r]);
                ap[r] = fmaf(ps[mm][r], u, ap[r]);
            }
        }
        __syncthreads();
    }
    #pragma unroll
    for (int r = 0; r < RANK; ++r) {
        QtU[((size_t)b * RANK + r) * IN_DIM + n] = aq[r];
        PtU[((size_t)b * RANK + r) * IN_DIM + n] = ap[r];
    }
}

// =====================================================================
// 5b) rank-16 update:
//   U[m,n] = clip(nz(U + (s/4)*(P(QtU) - Q(PtU))[m,n] + (s/4)*Bm[m-IN,n]))
// =====================================================================
__global__ void k_step_update(float* __restrict__ U,
                              const float* __restrict__ Pf, const float* __restrict__ Qf,
                              const float* __restrict__ Bpf, const float* __restrict__ Bqf,
                              const float* __restrict__ QtU, const float* __restrict__ PtU,
                              const float* __restrict__ sarr) {
    int b = blockIdx.y;
    int n = blockIdx.x * 256 + threadIdx.x;
    float s4 = sarr[b] * (1.0f / STEPS);
    float qtu[RANK], ptu[RANK], bq[RANK];
    #pragma unroll
    for (int r = 0; r < RANK; ++r) {
        qtu[r] = QtU[((size_t)b * RANK + r) * IN_DIM + n];
        ptu[r] = PtU[((size_t)b * RANK + r) * IN_DIM + n];
        bq[r]  = Bqf[((size_t)b * IN_DIM + n) * RANK + r];
    }
    __shared__ float ps[128][RANK];
    __shared__ float qs[128][RANK];
    __shared__ float bps[128][RANK];
    float* Ub = U + (size_t)b * OUT_DIM * IN_DIM;
    const float* Pb  = Pf  + (size_t)b * OUT_DIM * RANK;
    const float* Qb  = Qf  + (size_t)b * OUT_DIM * RANK;
    const float* Bpb = Bpf + (size_t)b * (OUT_DIM - IN_DIM) * RANK;
    for (int m0 = 0; m0 < OUT_DIM; m0 += 128) {
        for (int i = threadIdx.x; i < 128 * RANK; i += 256) {
            ps[i >> 3][i & 7] = Pb[m0 * RANK + i];
            qs[i >> 3][i & 7] = Qb[m0 * RANK + i];
            if (m0 >= IN_DIM) bps[i >> 3][i & 7] = Bpb[(m0 - IN_DIM) * RANK + i];
        }
        __syncthreads();
        for (int mm = 0; mm < 128; ++mm) {
            int m = m0 + mm;
            float u = Ub[(size_t)m * IN_DIM + n];
            float acc = 0.f;
            #pragma unroll
            for (int r = 0; r < RANK; ++r)
                acc += ps[mm][r] * qtu[r] - qs[mm][r] * ptu[r];
            if (m >= IN_DIM) {
                #pragma unroll
                for (int r = 0; r < RANK; ++r) acc = fmaf(bps[mm][r], bq[r], acc);
            }
            float v = fmaf(s4, acc, u);
            v = __builtin_isfinite(v) ? v : 0.f;          // nan_to_num
            v = fminf(fmaxf(v, -CLAMPV), CLAMPV);         // clip
            Ub[(size_t)m * IN_DIM + n] = v;
        }
        __syncthreads();
    }
}

// =====================================================================
// 6) final GEMM (compute-bound, f32 WMMA), async double-buffered:
//    out[b,n,m] = sum_d x[b,n,d]*U[b,m,d] + bias[m]
//    WG = 256 thr (8 waves), tile 64(n) x 128(m), K-chunk 32.
//    Waves 4x2; each wave 16(n) x 64(m) = 4 x V_WMMA_F32_16X16X4_F32 accums.
//    Staging uses GLOBAL_LOAD_ASYNC_TO_LDS_B128 into the shadow buffer
//    while wmma consumes the current one (ASYNCcnt + barrier flip).
// =====================================================================
#define GK   32
#define LDP  36            // row stride (floats): 144B = 16B-aligned, odd/4 -> bank spread
#define BN   64
#define BM   128
#define NCH  (IN_DIM / GK) // 32 chunks

__global__ __launch_bounds__(256)
void k_out_wmma(const float* __restrict__ x, const float* __restrict__ U,
                const float* __restrict__ bias, float* __restrict__ out) {
    __shared__ float xs[2][BN][LDP];   // x tile  [n][k]
    __shared__ float us[2][BM][LDP];   // U tile  [m][k]

    const int b    = blockIdx.z;
    const int m_wg = blockIdx.x * BM;
    const int n_wg = blockIdx.y * BN;

    const int tid  = threadIdx.x;
    const int lane = tid & 31;
    const int wid  = tid >> 5;          // 0..7
    const int wn   = wid >> 1;          // 0..3 -> n offset wn*16
    const int wm   = wid & 1;           // 0..1 -> m offset wm*64
    const int lr   = lane & 15;
    const int half = lane >> 4;         // selects K pair per ISA A/B layout
    const int kof  = half * 2;

    const float* xb = x + ((size_t)b * NSEQ    + n_wg) * IN_DIM;
    const float* Ub = U + ((size_t)b * OUT_DIM + m_wg) * IN_DIM;

    // per-thread staging coordinates (float4 granularity, 8 float4 per row)
    const int xr0 = tid >> 3;                  // rows 0..31   (it adds +32)
    const int ur0 = tid >> 3;                  // rows 0..31   (it adds +32 each)
    const int c4  = (tid & 7) << 2;

    const unsigned xs_lds[2] = { lds_addr_of(&xs[0][0][0]), lds_addr_of(&xs[1][0][0]) };
    const unsigned us_lds[2] = { lds_addr_of(&us[0][0][0]), lds_addr_of(&us[1][0][0]) };

    // ---- async stage of one K-chunk into buffer `bu` ----
    auto stage = [&](int bu, int k0) {
        #pragma unroll
        for (int it = 0; it < 2; ++it) {       // 64x32 floats of x
            int row = xr0 + it * 32;
            async_load_b128(xs_lds[bu] + (unsigned)(row * LDP + c4) * 4,
                            &xb[(size_t)row * IN_DIM + k0 + c4]);
        }
        #pragma unroll
        for (int it = 0; it < 4; ++it) {       // 128x32 floats of U
            int row = ur0 + it * 32;
            async_load_b128(us_lds[bu] + (unsigned)(row * LDP + c4) * 4,
                            &Ub[(size_t)row * IN_DIM + k0 + c4]);
        }
    };

    v8f acc[4];
    #pragma unroll
    for (int j = 0; j < 4; ++j) acc[j] = (v8f){0.f,0.f,0.f,0.f,0.f,0.f,0.f,0.f};

    const int an    = wn * 16 + lr;     // A-frag row in xs
    const int mbase = wm * 64 + lr;     // B-frag base row in us

    stage(0, 0);                        // prologue prefetch

    for (int i = 0; i < NCH; ++i) {
        const int cur = i & 1;
        wait_async0();                  // this wave's loads into buf[cur] done
        __syncthreads();                // everyone's loads done / prev reads done
        if (i + 1 < NCH) stage(cur ^ 1, (i + 1) * GK);

        #pragma unroll
        for (int kk = 0; kk < GK; kk += 4) {
            v2f a;
            a.x = xs[cur][an][kk + kof];
            a.y = xs[cur][an][kk + kof + 1];
            #pragma unroll
            for (int j = 0; j < 4; ++j) {
                const int mr = mbase + j * 16;
                v2f bb;
                bb.x = us[cur][mr][kk + kof];
                bb.y = us[cur][mr][kk + kof + 1];
                acc[j] = __builtin_amdgcn_wmma_f32_16x16x4_f32(
                    false, a, false, bb, (short)0, acc[j], false, false);
            }
        }
        __syncthreads();                // done reading buf[cur]; safe to overwrite next iter
    }

    // ---- epilogue: D layout -> vgpr v holds rows {v, v+8}, cols = lane%16 ----
    #pragma unroll
    for (int j = 0; j < 4; ++j) {
        const int mg = m_wg + wm * 64 + j * 16 + lr;
        const float bval = bias[mg];
        #pragma unroll
        for (int v = 0; v < 8; ++v) {
            const int ng = n_wg + wn * 16 + half * 8 + v;
            __builtin_nontemporal_store(acc[j][v] + bval,
                &out[((size_t)b * NSEQ + ng) * OUT_DIM + mg]);
        }
    }
}

// =====================================================================
// launcher
// =====================================================================
extern "C" void kernel_launch(void* const* d_in, const int* in_sizes, int n_in,
                              void* d_out, int out_size, void* d_ws, size_t ws_size,
                              hipStream_t stream) {
    const float* x    = (const float*)d_in[0];
    const float* W_ap = (const float*)d_in[1];
    const float* b_ap = (const float*)d_in[2];
    const float* W_aq = (const float*)d_in[3];
    const float* b_aq = (const float*)d_in[4];
    const float* W_bp = (const float*)d_in[5];
    const float* b_bp = (const float*)d_in[6];
    const float* W_bq = (const float*)d_in[7];
    const float* b_bq = (const float*)d_in[8];
    const float* U0   = (const float*)d_in[9];
    const float* bias = (const float*)d_in[10];
    float* out = (float*)d_out;

    // workspace layout (floats)
    float* ws = (float*)d_ws;
    size_t off = 0;
    float* U    = ws + off; off += (size_t)BATCH * OUT_DIM * IN_DIM;           // 64MB
    float* z    = ws + off; off += (size_t)BATCH * IN_DIM;
    float* P    = ws + off; off += (size_t)BATCH * OUT_DIM * RANK;
    float* Q    = ws + off; off += (size_t)BATCH * OUT_DIM * RANK;
    float* Bp   = ws + off; off += (size_t)BATCH * (OUT_DIM - IN_DIM) * RANK;
    float* Bq   = ws + off; off += (size_t)BATCH * IN_DIM * RANK;
    float* QtU  = ws + off; off += (size_t)BATCH * RANK * IN_DIM;
    float* PtU  = ws + off; off += (size_t)BATCH * RANK * IN_DIM;
    float* sarr = ws + off; off += BATCH;
    (void)ws_size; (void)n_in; (void)in_sizes; (void)out_size;

    // 1) z = mean(x)
    k_mean<<<dim3(IN_DIM / 256, BATCH), 256, 0, stream>>>(x, z);

    // 2) controller projections (NT weight streams)
    k_zmatmul<<<dim3((OUT_DIM * RANK) / 256, BATCH), 256, 0, stream>>>(z, W_ap, b_ap, P,  OUT_DIM * RANK);
    k_zmatmul<<<dim3((OUT_DIM * RANK) / 256, BATCH), 256, 0, stream>>>(z, W_aq, b_aq, Q,  OUT_DIM * RANK);
    k_zmatmul<<<dim3(((OUT_DIM - IN_DIM) * RANK) / 256, BATCH), 256, 0, stream>>>(z, W_bp, b_bp, Bp, (OUT_DIM - IN_DIM) * RANK);
    k_zmatmul<<<dim3((IN_DIM * RANK) / 256, BATCH), 256, 0, stream>>>(z, W_bq, b_bq, Bq, IN_DIM * RANK);

    // 3) clamp scale from Gram matrices (no dense A/G materialization)
    k_scale<<<BATCH, 320, 0, stream>>>(P, Q, Bp, Bq, sarr);

    // 4) U = broadcast(U0)
    k_initU<<<dim3((OUT_DIM * IN_DIM / 4) / 256, BATCH), 256, 0, stream>>>(U0, U);

    // 5) Lie-group Euler steps, exploiting rank-16 structure of A
    for (int s = 0; s < STEPS; ++s) {
        k_step_thin<<<dim3(IN_DIM / 128, BATCH), 128, 0, stream>>>(U, P, Q, QtU, PtU);
        k_step_update<<<dim3(IN_DIM / 256, BATCH), 256, 0, stream>>>(U, P, Q, Bp, Bq, QtU, PtU, sarr);
    }

    // 6) out = x @ U^T + bias   (WMMA f32, async double-buffered LDS)
    k_out_wmma<<<dim3(OUT_DIM / BM, NSEQ / BN, BATCH), 256, 0, stream>>>(x, U, bias, out);
}